// MultiHeadAttention_20710332301940
// MI455X (gfx1250) — compile-verified
//
#include <hip/hip_runtime.h>
#include <hip/hip_bf16.h>

// ---------------------------------------------------------------------------
// MHA forward for MI455X (gfx1250), fp32 end-to-end via V_WMMA_F32_16X16X4_F32
//   B=2, S=2048, D_MODEL=1024, H=16, DK=64
// ---------------------------------------------------------------------------

typedef __attribute__((ext_vector_type(2))) float v2f;
typedef __attribute__((ext_vector_type(8))) float v8f;

#define B_     2
#define S_     2048
#define DM_    1024
#define H_     16
#define DK_    64
#define BH_    (B_ * H_)
#define ROWS_  (B_ * S_)      // 4096
#define N_QKV  (3 * DM_)      // 3072

__device__ __forceinline__ v8f wmma_f32(v2f a, v2f b, v8f c) {
  // 8 args: (neg_a, A, neg_b, B, c_mod, C, reuse_a, reuse_b)
  return __builtin_amdgcn_wmma_f32_16x16x4_f32(
      false, a, false, b, (short)0, c, false, false);
}

// ---------------------------------------------------------------------------
// Shared GEMM core: C[128x64 tile] = A[M,K] * W[N,K]^T
// 256 threads = 8 waves; wave (wm in 0..3, wn in 0..1) owns a 32x32 patch
// LDS staging padded to 20 floats/row -> conflict-free fragment reads.
// ---------------------------------------------------------------------------
__device__ __forceinline__ void gemm_core_128x64(
    const float* __restrict__ A, const float* __restrict__ W,
    float* Xs, float* Ws, int m0, int n0, int Kdim, v8f acc[2][2]) {
  const int tid = threadIdx.x;
  const int wv = tid >> 5;
  const int lane = tid & 31;
  const int r = lane & 15;
  const int hh = lane >> 4;
  const int wm = wv >> 1;    // 0..3 (M)
  const int wn = wv & 1;     // 0..1 (N)

  for (int k0 = 0; k0 < Kdim; k0 += 16) {
    __syncthreads();
    // Stage X tile: 128 rows x 16 cols (512 float4, 2 per thread)
#pragma unroll
    for (int i = 0; i < 2; ++i) {
      int idx = tid + i * 256;
      int row = idx >> 2;
      int cv = (idx & 3) * 4;
      *(float4*)(Xs + row * 20 + cv) =
          *(const float4*)(A + (size_t)(m0 + row) * Kdim + k0 + cv);
    }
    // Stage W tile: 64 rows x 16 cols (256 float4, 1 per thread)
    {
      int row = tid >> 2;
      int cv = (tid & 3) * 4;
      *(float4*)(Ws + row * 20 + cv) =
          *(const float4*)(W + (size_t)(n0 + row) * Kdim + k0 + cv);
    }
    __syncthreads();
#pragma unroll
    for (int kk = 0; kk < 4; ++kk) {
      const int kd = kk * 4 + hh * 2;
      const float* xr0 = Xs + (wm * 32 + r) * 20 + kd;
      const float* xr1 = Xs + (wm * 32 + 16 + r) * 20 + kd;
      const float* wr0 = Ws + (wn * 32 + r) * 20 + kd;
      const float* wr1 = Ws + (wn * 32 + 16 + r) * 20 + kd;
      v2f a0 = {xr0[0], xr0[1]};
      v2f a1 = {xr1[0], xr1[1]};
      v2f b0 = {wr0[0], wr0[1]};
      v2f b1 = {wr1[0], wr1[1]};
      acc[0][0] = wmma_f32(a0, b0, acc[0][0]);
      acc[0][1] = wmma_f32(a0, b1, acc[0][1]);
      acc[1][0] = wmma_f32(a1, b0, acc[1][0]);
      acc[1][1] = wmma_f32(a1, b1, acc[1][1]);
    }
  }
}

// ---------------------------------------------------------------------------
// Kernel 1: QKV projection + bias + RoPE, scatter to (B,H,S,DK) Q/K/V buffers
// grid = (ROWS_/128, N_QKV/64), block = 256
// ---------------------------------------------------------------------------
__global__ void __launch_bounds__(256)
qkv_rope_kernel(const float* __restrict__ x, const float* __restrict__ w_qkv,
                const float* __restrict__ b_qkv, float* __restrict__ Qw,
                float* __restrict__ Kw, float* __restrict__ Vw) {
  __shared__ float Xs[128 * 20];
  __shared__ float Ws[64 * 20];
  v8f acc[2][2] = {};
  const int m0 = blockIdx.x * 128;
  const int n0 = blockIdx.y * 64;
  gemm_core_128x64(x, w_qkv, Xs, Ws, m0, n0, DM_, acc);

  const int tid = threadIdx.x;
  const int wv = tid >> 5;
  const int lane = tid & 31;
  const int r = lane & 15;
  const int hhalf = lane >> 4;
  const int wm = wv >> 1;
  const int wn = wv & 1;

#pragma unroll
  for (int mi = 0; mi < 2; ++mi) {
#pragma unroll
    for (int ni = 0; ni < 2; ++ni) {
      const int o = n0 + wn * 32 + ni * 16 + r;  // global out-feature
      const int sel = o >> 10;                   // 0=q,1=k,2=v (uniform per tile)
      const int rem = o & 1023;
      const int head = rem >> 6;
      const int d = rem & 63;
      const float bv = b_qkv[o];
      // inv_freq = 10000^(-(2p)/64) = exp(-(2p)*ln(10000)/64)
      const int p = d >> 1;
      const float invf = __expf((float)(2 * p) * (-9.210340371976184f / 64.0f));
      const float sgn = (d & 1) ? 1.0f : -1.0f;
      float* dst = (sel == 0) ? Qw : ((sel == 1) ? Kw : Vw);
#pragma unroll
      for (int g = 0; g < 8; ++g) {
        const int gm = m0 + wm * 32 + mi * 16 + g + 8 * hhalf;  // global row
        const int srow = gm & (S_ - 1);
        const int bb = gm >> 11;
        float v = acc[mi][ni][g] + bv;
        float other = __shfl_xor(v, 1);  // RoPE partner (even<->odd d)
        float outv = v;
        if (sel < 2) {
          const float ang = (float)srow * invf;
          const float sn = __sinf(ang);
          const float cs = __cosf(ang);
          outv = v * cs + other * sgn * sn;
        }
        dst[(((size_t)(bb * H_ + head)) * S_ + srow) * DK_ + d] = outv;
      }
    }
  }
}

// ---------------------------------------------------------------------------
// Kernel 2: flash attention with causal mask.
// grid = (S_/128, BH_), block = 256 (8 waves x 16 query rows)
// ---------------------------------------------------------------------------
__global__ void __launch_bounds__(256)
flash_attn_kernel(const float* __restrict__ Q, const float* __restrict__ K,
                  const float* __restrict__ V, float* __restrict__ O) {
  __shared__ float Ks[16 * 68];       // padded: 68 floats/row
  __shared__ float Vs[16 * 68];
  __shared__ float Ps[8 * 16 * 17];   // per-wave P staging, padded 17/row

  const int bh = blockIdx.y;
  const int qb = blockIdx.x;
  const int tid = threadIdx.x;
  const int wv = tid >> 5;
  const int lane = tid & 31;
  const int r = lane & 15;
  const int hhalf = lane >> 4;
  const int qw0 = qb * 128 + wv * 16;  // first query row for this wave

  const float* Qp = Q + (size_t)bh * S_ * DK_;
  const float* Kp = K + (size_t)bh * S_ * DK_;
  const float* Vp = V + (size_t)bh * S_ * DK_;

  // Q fragments (16 k-steps of K=4), pre-scaled by 1/sqrt(dk)
  v2f qf[16];
#pragma unroll
  for (int kk = 0; kk < 16; ++kk) {
    const float* qp = Qp + (size_t)(qw0 + r) * DK_ + kk * 4 + hhalf * 2;
    qf[kk] = v2f{qp[0] * 0.125f, qp[1] * 0.125f};
  }

  v8f o0 = {}, o1 = {}, o2 = {}, o3 = {};
  float mrow[8], lrow[8];
#pragma unroll
  for (int g = 0; g < 8; ++g) { mrow[g] = -3.0e38f; lrow[g] = 0.0f; }

  float* Pw = Ps + wv * 16 * 17;
  const int njt = qb * 8 + 8;  // k-tiles needed by the last wave in the block

  for (int j = 0; j < njt; ++j) {
    __syncthreads();
    {  // cooperative K/V tile load: 16 rows x 64 cols, one float4 each
      const int rowj = tid >> 4;
      const int cv = (tid & 15) * 4;
      const float4 kv = *(const float4*)(Kp + (size_t)(j * 16 + rowj) * DK_ + cv);
      const float4 vvv = *(const float4*)(Vp + (size_t)(j * 16 + rowj) * DK_ + cv);
      *(float4*)(Ks + rowj * 68 + cv) = kv;
      *(float4*)(Vs + rowj * 68 + cv) = vvv;
    }
    __syncthreads();

    if (j * 16 <= qw0 + 15) {  // tile intersects this wave's causal range
      // S = Q * K^T  (16x16)
      v8f s = {};
#pragma unroll
      for (int kk = 0; kk < 16; ++kk) {
        const int kd = kk * 4 + hhalf * 2;
        const float* kr = Ks + r * 68 + kd;
        v2f bf = {kr[0], kr[1]};
        s = wmma_f32(qf[kk], bf, s);
      }
      // causal mask on the diagonal tile
      if (j * 16 + 15 > qw0) {
        const int key = j * 16 + r;
#pragma unroll
        for (int g = 0; g < 8; ++g) {
          if (key > qw0 + g + 8 * hhalf) s[g] = -3.0e38f;
        }
      }
      // online softmax (row = across lanes 0-15 of each half)
      float pv[8];
#pragma unroll
      for (int g = 0; g < 8; ++g) {
        float v = s[g];
        float mx = v;
        mx = fmaxf(mx, __shfl_xor(mx, 1));
        mx = fmaxf(mx, __shfl_xor(mx, 2));
        mx = fmaxf(mx, __shfl_xor(mx, 4));
        mx = fmaxf(mx, __shfl_xor(mx, 8));
        const float mnew = fmaxf(mrow[g], mx);
        const float alpha = __expf(mrow[g] - mnew);
        const float pe = __expf(v - mnew);
        float rs = pe;
        rs += __shfl_xor(rs, 1);
        rs += __shfl_xor(rs, 2);
        rs += __shfl_xor(rs, 4);
        rs += __shfl_xor(rs, 8);
        lrow[g] = lrow[g] * alpha + rs;
        mrow[g] = mnew;
        pv[g] = pe;
        o0[g] *= alpha; o1[g] *= alpha; o2[g] *= alpha; o3[g] *= alpha;
      }
      // reshape P: C-layout -> A-layout via wave-private LDS patch
#pragma unroll
      for (int g = 0; g < 8; ++g) Pw[(g + 8 * hhalf) * 17 + r] = pv[g];
      // O += P * V
#pragma unroll
      for (int kk2 = 0; kk2 < 4; ++kk2) {
        const int kd = kk2 * 4 + hhalf * 2;
        const float* pr = Pw + r * 17 + kd;
        v2f af = {pr[0], pr[1]};
        v2f b0 = {Vs[kd * 68 + 0 * 16 + r], Vs[(kd + 1) * 68 + 0 * 16 + r]};
        v2f b1 = {Vs[kd * 68 + 1 * 16 + r], Vs[(kd + 1) * 68 + 1 * 16 + r]};
        v2f b2 = {Vs[kd * 68 + 2 * 16 + r], Vs[(kd + 1) * 68 + 2 * 16 + r]};
        v2f b3 = {Vs[kd * 68 + 3 * 16 + r], Vs[(kd + 1) * 68 + 3 * 16 + r]};
        o0 = wmma_f32(af, b0, o0);
        o1 = wmma_f32(af, b1, o1);
        o2 = wmma_f32(af, b2, o2);
        o3 = wmma_f32(af, b3, o3);
      }
    }
  }

  // normalize (ref: ex / (sum + 1e-8)) and store to (B,S,H*DK)
  const int bb = bh >> 4;
  const int head = bh & 15;
#pragma unroll
  for (int g = 0; g < 8; ++g) {
    const float inv = 1.0f / (lrow[g] + 1e-8f);
    const int qrow = qw0 + g + 8 * hhalf;
    float* dst = O + ((size_t)(bb * S_ + qrow)) * DM_ + head * DK_;
    dst[0 * 16 + r] = o0[g] * inv;
    dst[1 * 16 + r] = o1[g] * inv;
    dst[2 * 16 + r] = o2[g] * inv;
    dst[3 * 16 + r] = o3[g] * inv;
  }
}

// ---------------------------------------------------------------------------
// Kernel 3: output projection + bias -> d_out (B,S,DM)
// grid = (ROWS_/128, DM_/64), block = 256
// ---------------------------------------------------------------------------
__global__ void __launch_bounds__(256)
out_proj_kernel(const float* __restrict__ Oin, const float* __restrict__ w_out,
                const float* __restrict__ b_out, float* __restrict__ out) {
  __shared__ float Xs[128 * 20];
  __shared__ float Ws[64 * 20];
  v8f acc[2][2] = {};
  const int m0 = blockIdx.x * 128;
  const int n0 = blockIdx.y * 64;
  gemm_core_128x64(Oin, w_out, Xs, Ws, m0, n0, DM_, acc);

  const int tid = threadIdx.x;
  const int wv = tid >> 5;
  const int lane = tid & 31;
  const int r = lane & 15;
  const int hhalf = lane >> 4;
  const int wm = wv >> 1;
  const int wn = wv & 1;

#pragma unroll
  for (int mi = 0; mi < 2; ++mi) {
#pragma unroll
    for (int ni = 0; ni < 2; ++ni) {
      const int n = n0 + wn * 32 + ni * 16 + r;
      const float bv = b_out[n];
#pragma unroll
      for (int g = 0; g < 8; ++g) {
        const int gm = m0 + wm * 32 + mi * 16 + g + 8 * hhalf;
        out[(size_t)gm * DM_ + n] = acc[mi][ni][g] + bv;
      }
    }
  }
}

// ---------------------------------------------------------------------------
extern "C" void kernel_launch(void* const* d_in, const int* in_sizes, int n_in,
                              void* d_out, int out_size, void* d_ws,
                              size_t ws_size, hipStream_t stream) {
  const float* x = (const float*)d_in[0];
  const float* w_qkv = (const float*)d_in[1];
  const float* b_qkv = (const float*)d_in[2];
  const float* w_out = (const float*)d_in[3];
  const float* b_out = (const float*)d_in[4];
  float* out = (float*)d_out;

  const size_t per_buf = (size_t)BH_ * S_ * DK_;  // 4M floats = 16 MB
  float* Qw = (float*)d_ws;
  float* Kw = Qw + per_buf;
  float* Vw = Kw + per_buf;
  float* Ow = Vw + per_buf;  // (B,S,DM) = 4M floats

  qkv_rope_kernel<<<dim3(ROWS_ / 128, N_QKV / 64), 256, 0, stream>>>(
      x, w_qkv, b_qkv, Qw, Kw, Vw);
  flash_attn_kernel<<<dim3(S_ / 128, BH_), 256, 0, stream>>>(Qw, Kw, Vw, Ow);
  out_proj_kernel<<<dim3(ROWS_ / 128, DM_ / 64), 256, 0, stream>>>(
      Ow, w_out, b_out, out);
}